// LoTDBatched_13537736917477
// MI455X (gfx1250) — compile-verified
//
#include <hip/hip_runtime.h>
#include <stdint.h>

// Instant-NGP style batched hash-grid encode for MI455X (gfx1250).
//
// Key hardware insight: MI455X has a 192 MB global L2. Processed in natural
// order, the fine hash levels touch 256 MB of tables randomly -> HBM-bound
// (~3.6 GB effective). Processed grouped by batch index (16-bin counting
// sort), the concurrently-hot set is a few 32 MiB batch slices + 48 MB of
// coarse levels, which fits in L2 -> params are pulled from HBM ~once
// (~0.8 GB total traffic, ~4-5x less).

#define LVL    8
#define FEAT   2
#define TBITS  19
#define TMASK  ((1u << TBITS) - 1u)
#define P1     2654435761u
#define P2     805459861u
#define NB     16

typedef __attribute__((ext_vector_type(2))) float f32x2;
typedef __attribute__((ext_vector_type(4))) float f32x4;

// ---- workspace layout (uint32 words) ----
// [0..15]   histogram
// [16..31]  scatter cursors
// [32..48]  bin offsets (unused by later kernels, kept for debug)
// byte 256+ permutation (N uint32)

__global__ void ws_zero_kernel(unsigned int* ws) {
    if (threadIdx.x < 64) ws[threadIdx.x] = 0u;
}

__global__ __launch_bounds__(256) void hist_kernel(
    const int* __restrict__ bidx, int n, unsigned int* __restrict__ hist)
{
    __shared__ unsigned int lh[NB];
    if (threadIdx.x < NB) lh[threadIdx.x] = 0u;
    __syncthreads();
    const int i = blockIdx.x * blockDim.x + threadIdx.x;
    if (i < n) atomicAdd(&lh[bidx[i] & (NB - 1)], 1u);
    __syncthreads();
    if (threadIdx.x < NB) atomicAdd(&hist[threadIdx.x], lh[threadIdx.x]);
}

__global__ void scan_kernel(unsigned int* ws) {
    if (threadIdx.x == 0) {
        unsigned int acc = 0u;
        for (int b = 0; b < NB; ++b) {
            ws[32 + b] = acc;   // offsets
            ws[16 + b] = acc;   // cursors
            acc += ws[b];
        }
        ws[32 + NB] = acc;
    }
}

__global__ __launch_bounds__(256) void scatter_kernel(
    const int* __restrict__ bidx, int n,
    unsigned int* __restrict__ cursors, unsigned int* __restrict__ perm)
{
    const int i = blockIdx.x * blockDim.x + threadIdx.x;
    if (i < n) {
        const int b = bidx[i] & (NB - 1);
        const unsigned int pos = atomicAdd(&cursors[b], 1u);
        perm[pos] = (unsigned int)i;
    }
}

template <bool SORTED>
__global__ __launch_bounds__(256) void lotd_encode_kernel(
    const float* __restrict__ inp,     // N x 3, in [-1,1]
    const float* __restrict__ params,  // B x L x T x F
    const int*   __restrict__ bidx,    // N
    const unsigned int* __restrict__ perm,
    float*       __restrict__ out,     // N x (L*F)
    int nPts)
{
    const int i = blockIdx.x * blockDim.x + threadIdx.x;
    if (i >= nPts) return;
    const unsigned int j = SORTED ? perm[i] : (unsigned int)i;

    const float res[LVL] = {16.f, 26.f, 43.f, 71.f, 116.f, 190.f, 312.f, 512.f};

    // ---- input ----
    const float* ip = inp + 3ull * j;
    float x, y, z;
    int b;
    if (SORTED) {
        // gathered via perm; input (24 MB) stays L2-resident -> regular TH
        x = fmaf(ip[0], 0.5f, 0.5f);
        y = fmaf(ip[1], 0.5f, 0.5f);
        z = fmaf(ip[2], 0.5f, 0.5f);
        b = bidx[j];
    } else {
        // linear streaming read-once -> NT
        x = fmaf(__builtin_nontemporal_load(ip + 0), 0.5f, 0.5f);
        y = fmaf(__builtin_nontemporal_load(ip + 1), 0.5f, 0.5f);
        z = fmaf(__builtin_nontemporal_load(ip + 2), 0.5f, 0.5f);
        b = __builtin_nontemporal_load(bidx + j);
    }

    // per-lane table base: params + b * (L*T*F) floats (32 MiB per batch slice)
    const char* tab = (const char*)(params)
                    + (unsigned long long)(b & (NB - 1))
                      * (LVL * (1ull << TBITS) * FEAT * sizeof(float));

    // ---- phase 1: compute all 64 indices, issue all 64 gathers ----
    f32x2 g[LVL][8];
    #pragma unroll
    for (int l = 0; l < LVL; ++l) {
        const float R = res[l];
        const uint32_t xi = (uint32_t)(x * R);
        const uint32_t yi = (uint32_t)(y * R);
        const uint32_t zi = (uint32_t)(z * R);

        const uint32_t hx0 = xi,      hx1 = xi + 1u;
        const uint32_t hy0 = yi * P1, hy1 = (yi + 1u) * P1;
        const uint32_t hz0 = zi * P2, hz1 = (zi + 1u) * P2;
        const uint32_t e00 = hy0 ^ hz0, e01 = hy0 ^ hz1;
        const uint32_t e10 = hy1 ^ hz0, e11 = hy1 ^ hz1;

        uint32_t idx[8];
        idx[0] = (hx0 ^ e00) & TMASK;
        idx[1] = (hx0 ^ e01) & TMASK;
        idx[2] = (hx0 ^ e10) & TMASK;
        idx[3] = (hx0 ^ e11) & TMASK;
        idx[4] = (hx1 ^ e00) & TMASK;
        idx[5] = (hx1 ^ e01) & TMASK;
        idx[6] = (hx1 ^ e10) & TMASK;
        idx[7] = (hx1 ^ e11) & TMASK;

        const unsigned long long lbase = (unsigned long long)l << 22; // l*T*F*4
        #pragma unroll
        for (int k = 0; k < 8; ++k) {
            const f32x2* p = (const f32x2*)(tab + lbase + ((unsigned long long)idx[k] << 3));
            if (!SORTED && l >= 4) {
                // unsorted: fine levels exceed L2 -> NT protects coarse levels
                g[l][k] = __builtin_nontemporal_load(p);
            } else {
                // sorted: hot batch slices fit in 192 MB L2 -> regular TH
                g[l][k] = *p;
            }
        }
    }

    // ---- phase 2: trilinear weights + accumulate (loads drain meanwhile) ----
    float acc[LVL * FEAT];
    #pragma unroll
    for (int l = 0; l < LVL; ++l) {
        const float R  = res[l];
        const float px = x * R, py = y * R, pz = z * R;
        const float fx = px - floorf(px);
        const float fy = py - floorf(py);
        const float fz = pz - floorf(pz);
        const float wx[2] = {1.f - fx, fx};
        const float wy[2] = {1.f - fy, fy};
        const float wz[2] = {1.f - fz, fz};

        float f0 = 0.f, f1 = 0.f;
        #pragma unroll
        for (int k = 0; k < 8; ++k) {
            const float w = wx[(k >> 2) & 1] * wy[(k >> 1) & 1] * wz[k & 1];
            f0 = fmaf(w, g[l][k].x, f0);
            f1 = fmaf(w, g[l][k].y, f1);
        }
        acc[2 * l + 0] = f0;
        acc[2 * l + 1] = f1;
    }

    // ---- output: never re-read -> NT b128 stores, 64 B per lane ----
    f32x4* op = (f32x4*)(out + (unsigned long long)j * (LVL * FEAT));
    #pragma unroll
    for (int q = 0; q < 4; ++q) {
        f32x4 v = {acc[4 * q + 0], acc[4 * q + 1], acc[4 * q + 2], acc[4 * q + 3]};
        __builtin_nontemporal_store(v, op + q);
    }
}

extern "C" void kernel_launch(void* const* d_in, const int* in_sizes, int n_in,
                              void* d_out, int out_size, void* d_ws, size_t ws_size,
                              hipStream_t stream) {
    const float* inp    = (const float*)d_in[0];   // N*3
    const float* params = (const float*)d_in[1];   // B*L*T*F
    const int*   bidx   = (const int*)d_in[2];     // N
    float*       out    = (float*)d_out;           // N*16

    const int nPts   = in_sizes[0] / 3;
    const int threads = 256;
    const int blocks  = (nPts + threads - 1) / threads;

    const size_t wsNeeded = 256ull + 4ull * (size_t)nPts;
    if (ws_size >= wsNeeded && d_ws != nullptr) {
        unsigned int* ws   = (unsigned int*)d_ws;
        unsigned int* perm = (unsigned int*)((char*)d_ws + 256);
        ws_zero_kernel<<<1, 64, 0, stream>>>(ws);
        hist_kernel<<<blocks, threads, 0, stream>>>(bidx, nPts, ws);
        scan_kernel<<<1, 32, 0, stream>>>(ws);
        scatter_kernel<<<blocks, threads, 0, stream>>>(bidx, nPts, ws + 16, perm);
        lotd_encode_kernel<true><<<blocks, threads, 0, stream>>>(
            inp, params, bidx, perm, out, nPts);
    } else {
        lotd_encode_kernel<false><<<blocks, threads, 0, stream>>>(
            inp, params, bidx, nullptr, out, nPts);
    }
}